// EnhancedGNNJobRecommender_60662118088918
// MI455X (gfx1250) — compile-verified
//
#include <hip/hip_runtime.h>

// ---------------- problem constants ----------------
#define NJOBS   2000
#define NNODES  50000
#define NEDGES  400000
#define HIDD    256
#define NEGS    0.2f
#define BNEPS   1e-5f
#define SMEPS   1e-16f

typedef __attribute__((ext_vector_type(2))) float v2f;
typedef __attribute__((ext_vector_type(4))) float v4f;
typedef __attribute__((ext_vector_type(8))) float v8f;
typedef __attribute__((ext_vector_type(4))) int   v4i;

#define AS1 __attribute__((address_space(1)))
#define AS3 __attribute__((address_space(3)))

#if defined(__gfx1250__) && __has_builtin(__builtin_amdgcn_global_load_async_to_lds_b128)
#define USE_ASYNC 1
#else
#define USE_ASYNC 0
#endif

__device__ __forceinline__ void wait_async_le(int n) {
#if USE_ASYNC
#if __has_builtin(__builtin_amdgcn_s_wait_asynccnt)
    if (n == 0) __builtin_amdgcn_s_wait_asynccnt(0);
    else        __builtin_amdgcn_s_wait_asynccnt(1);
#else
    if (n == 0) asm volatile("s_wait_asynccnt 0x0" ::: "memory");
    else        asm volatile("s_wait_asynccnt 0x1" ::: "memory");
#endif
#endif
}

__device__ __forceinline__ float leakyf(float x) { return x >= 0.f ? x : NEGS * x; }
// order-preserving float<->int map for atomicMax on signed ints
__device__ __forceinline__ int   f2o(float f) { int i = __float_as_int(f); return i >= 0 ? i : (i ^ 0x7fffffff); }
__device__ __forceinline__ float o2f(int o)   { int i = o >= 0 ? o : (o ^ 0x7fffffff); return __int_as_float(i); }

// ---------------- generic fp32 WMMA GEMM: C = act(A[MxK] @ W[KxN] + b) ----------------
// one wave computes one 16x16 tile; M,N multiples of 16, K multiple of 4.
__global__ void wmma_gemm(const float* __restrict__ A, const float* __restrict__ W,
                          const float* __restrict__ bias, float* __restrict__ C,
                          int M, int N, int K, int act)
{
    int lane  = threadIdx.x & 31;
    int wave  = blockIdx.x * (blockDim.x >> 5) + (threadIdx.x >> 5);
    int mt_n  = M >> 4, nt_n = N >> 4;
    if (wave >= mt_n * nt_n) return;               // wave-uniform: EXEC stays all-ones
    int mt = wave / nt_n, nt = wave - mt * nt_n;
    int half = lane >> 4, l = lane & 15;

    const float* Ap = A + (size_t)(mt * 16 + l) * K + 2 * half;   // A: row=l, VGPR->K, halves K+2
    const float* Bp = W + (size_t)(2 * half) * N + nt * 16 + l;   // B: row=K, lanes->N

    v8f acc = {};
    for (int k = 0; k < K; k += 4) {
        v2f a = *(const v2f*)(Ap + k);
        v2f b;
        b.x = Bp[(size_t)k * N];
        b.y = Bp[(size_t)(k + 1) * N];
        acc = __builtin_amdgcn_wmma_f32_16x16x4_f32(false, a, false, b, (short)0, acc, false, false);
    }
    float bv = bias ? bias[nt * 16 + l] : 0.f;
    float* Cp = C + (size_t)(mt * 16 + 8 * half) * N + nt * 16 + l;
#pragma unroll
    for (int r = 0; r < 8; ++r) {
        float v = acc[r] + bv;
        if (act) v = leakyf(v);
        Cp[(size_t)r * N] = v;
    }
}

// ---------------- fused Q/K/V/skip projection with async-LDS A staging ----------------
// block = one 16-row M tile; 8 waves, each wave owns 2 N-tiles x 4 output matrices.
// A chunks (16x64) are DMA'd global->LDS (double buffered), fragments read via ds_load.
#define WMMA4(a_, b_, acc_) \
    acc_ = __builtin_amdgcn_wmma_f32_16x16x4_f32(false, a_, false, b_, (short)0, acc_, false, false)

__global__ void __launch_bounds__(256) wmma_gemm4_lds(
        const float* __restrict__ A,
        const float* __restrict__ Wq, const float* __restrict__ bq,
        const float* __restrict__ Wk, const float* __restrict__ bk,
        const float* __restrict__ Wv, const float* __restrict__ bv,
        const float* __restrict__ Ws, const float* __restrict__ bs,
        float* __restrict__ Q, float* __restrict__ Ko,
        float* __restrict__ V, float* __restrict__ S, int K)
{
    __shared__ float lA[2][16 * 64];                 // 2 x 4KB A chunks

    const int N  = HIDD;
    int t    = threadIdx.x;
    int mt   = blockIdx.x;
    int wv   = t >> 5;
    int lane = t & 31;
    int half = lane >> 4, l = lane & 15;

    // staging role: thread t copies 4 floats of the 16x64 chunk
    int sr = t >> 4, sc = (t & 15) << 2;
    const float* gA = A + (size_t)(mt * 16 + sr) * K + sc;
    float* lp0 = &lA[0][sr * 64 + sc];
    float* lp1 = &lA[1][sr * 64 + sc];

    auto stage = [&](int buf, int kc) {
        float* lp = buf ? lp1 : lp0;
#if USE_ASYNC
        __builtin_amdgcn_global_load_async_to_lds_b128(
            (AS1 v4i*)(gA + kc), (AS3 v4i*)lp, 0, 0);
#else
        *(v4f*)lp = *(const v4f*)(gA + kc);
#endif
    };

    int nt0 = wv * 2;
    int col = nt0 * 16 + l;
    const float* Bq = Wq + (size_t)(2 * half) * N + col;
    const float* Bk = Wk + (size_t)(2 * half) * N + col;
    const float* Bv = Wv + (size_t)(2 * half) * N + col;
    const float* Bs = Ws + (size_t)(2 * half) * N + col;

    v8f aq0 = {}, aq1 = {}, ak0 = {}, ak1 = {}, av0 = {}, av1 = {}, as0 = {}, as1 = {};

    int nchunks = K >> 6;
    stage(0, 0);
    for (int ch = 0; ch < nchunks; ++ch) {
        int buf = ch & 1;
        if (ch + 1 < nchunks) stage(buf ^ 1, (ch + 1) << 6);
        wait_async_le((ch + 1 < nchunks) ? 1 : 0);   // oldest (current) chunk landed
        __syncthreads();
        const float* la = &lA[buf][(l << 6) + (half << 1)];
        int kg = ch << 6;
#pragma unroll 4
        for (int k = 0; k < 64; k += 4) {
            v2f a = *(const v2f*)(la + k);           // ds_load_b64
            size_t ro = (size_t)(kg + k) * N;
            v2f b;
            b.x = Bq[ro];      b.y = Bq[ro + N];      WMMA4(a, b, aq0);
            b.x = Bq[ro + 16]; b.y = Bq[ro + N + 16]; WMMA4(a, b, aq1);
            b.x = Bk[ro];      b.y = Bk[ro + N];      WMMA4(a, b, ak0);
            b.x = Bk[ro + 16]; b.y = Bk[ro + N + 16]; WMMA4(a, b, ak1);
            b.x = Bv[ro];      b.y = Bv[ro + N];      WMMA4(a, b, av0);
            b.x = Bv[ro + 16]; b.y = Bv[ro + N + 16]; WMMA4(a, b, av1);
            b.x = Bs[ro];      b.y = Bs[ro + N];      WMMA4(a, b, as0);
            b.x = Bs[ro + 16]; b.y = Bs[ro + N + 16]; WMMA4(a, b, as1);
        }
        __syncthreads();                             // block done with this buffer
    }

    auto store2 = [&](float* C, const float* bias, v8f x0, v8f x1) {
        float b0 = bias[col], b1 = bias[col + 16];
        float* Cp = C + (size_t)(mt * 16 + 8 * half) * N + col;
#pragma unroll
        for (int r = 0; r < 8; ++r) {
            Cp[(size_t)r * N]      = x0[r] + b0;
            Cp[(size_t)r * N + 16] = x1[r] + b1;
        }
    };
    store2(Q,  bq, aq0, aq1);
    store2(Ko, bk, ak0, ak1);
    store2(V,  bv, av0, av1);
    store2(S,  bs, as0, as1);                        // skip term; attention scatters on top
}

// ---------------- per-edge attention ----------------
__global__ void edge_alpha_kernel(const int* __restrict__ ei, const float* __restrict__ ea,
                                  const float* __restrict__ We, const float* __restrict__ Q,
                                  const float* __restrict__ Kf, float* __restrict__ alpha,
                                  int* __restrict__ amax)
{
    int id = blockIdx.x * blockDim.x + threadIdx.x;
    if (id >= NEDGES * 4) return;
    int e = id >> 2, h = id & 3;
    int s = ei[e], d = ei[NEDGES + e];
    float a0 = ea[e * 3 + 0], a1 = ea[e * 3 + 1], a2 = ea[e * 3 + 2];
    const v4f* qp = (const v4f*)(Q  + (size_t)d * HIDD + h * 64);
    const v4f* kp = (const v4f*)(Kf + (size_t)s * HIDD + h * 64);
    const v4f* w0 = (const v4f*)(We + h * 64);
    const v4f* w1 = (const v4f*)(We + h * 64 + 256);
    const v4f* w2 = (const v4f*)(We + h * 64 + 512);
    v4f acc4 = {};
#pragma unroll 4
    for (int i = 0; i < 16; ++i) {
        v4f ev = a0 * w0[i] + a1 * w1[i] + a2 * w2[i];
        acc4 += qp[i] * (kp[i] + ev);
    }
    float acc = (acc4.x + acc4.y + acc4.z + acc4.w) * 0.125f;   // 1/sqrt(64)
    alpha[id] = acc;
    atomicMax(amax + d * 4 + h, f2o(acc));
}

__global__ void edge_ex_kernel(const int* __restrict__ ei, float* __restrict__ alpha,
                               const int* __restrict__ amax, float* __restrict__ denom)
{
    int id = blockIdx.x * blockDim.x + threadIdx.x;
    if (id >= NEDGES * 4) return;
    int e = id >> 2, h = id & 3;
    int d = ei[NEDGES + e];
    float ex = expf(alpha[id] - o2f(amax[d * 4 + h]));
    alpha[id] = ex;
    atomicAdd(denom + d * 4 + h, ex);
}

// one block per edge, 256 lanes = one feature each; e-projection recomputed on the fly
__global__ void edge_agg_kernel(const int* __restrict__ ei, const float* __restrict__ ea,
                                const float* __restrict__ We, const float* __restrict__ V,
                                const float* __restrict__ ex, const float* __restrict__ denom,
                                float* __restrict__ out)
{
    int e = blockIdx.x, j = threadIdx.x, h = j >> 6;
    int s = ei[e], d = ei[NEDGES + e];
    float w = ex[e * 4 + h] / (denom[d * 4 + h] + SMEPS);
    float a0 = ea[e * 3 + 0], a1 = ea[e * 3 + 1], a2 = ea[e * 3 + 2];
    float ev = a0 * We[j] + a1 * We[256 + j] + a2 * We[512 + j];
    atomicAdd(out + (size_t)d * HIDD + j, w * (V[(size_t)s * HIDD + j] + ev));
}

// ---------------- leaky + BatchNorm ----------------
__global__ void bn_reduce_kernel(const float* __restrict__ X, float* __restrict__ colsum,
                                 float* __restrict__ colsq, int M)
{
    int c = threadIdx.x;                       // 256 columns
    int per = (M + gridDim.x - 1) / gridDim.x;
    int r0 = blockIdx.x * per;
    int r1 = min(M, r0 + per);
    float s = 0.f, ss = 0.f;
    for (int r = r0; r < r1; ++r) {
        float y = leakyf(X[(size_t)r * HIDD + c]);
        s += y; ss += y * y;
    }
    atomicAdd(colsum + c, s);
    atomicAdd(colsq + c, ss);
}

__global__ void bn_apply_kernel(const float* __restrict__ X, const float* __restrict__ colsum,
                                const float* __restrict__ colsq, const float* __restrict__ g,
                                const float* __restrict__ b, float* __restrict__ Y, int M)
{
    int id = blockIdx.x * blockDim.x + threadIdx.x;
    if (id >= M * HIDD) return;
    int c = id & (HIDD - 1);
    float inv = 1.f / (float)M;
    float m   = colsum[c] * inv;
    float var = colsq[c] * inv - m * m;
    float ri  = rsqrtf(var + BNEPS);
    Y[id] = (leakyf(X[id]) - m) * ri * g[c] + b[c];
}

// ---------------- misc ----------------
__global__ void fill_i32(int* p, int v, int n)
{
    int i = blockIdx.x * blockDim.x + threadIdx.x;
    if (i < n) p[i] = v;
}

__global__ void dot_sigmoid_kernel(const float* __restrict__ X, const float* __restrict__ w,
                                   const float* __restrict__ b, float* __restrict__ out,
                                   int M, int K4)
{
    int r = blockIdx.x * blockDim.x + threadIdx.x;
    if (r >= M) return;
    const v4f* xp = (const v4f*)(X + (size_t)r * (K4 * 4));
    const v4f* wp = (const v4f*)w;
    v4f a4 = {};
    for (int i = 0; i < K4; ++i) a4 += xp[i] * wp[i];
    float acc = a4.x + a4.y + a4.z + a4.w + b[0];
    out[r] = 1.f / (1.f + expf(-acc));
}

// ---------------- host orchestration ----------------
extern "C" void kernel_launch(void* const* d_in, const int* in_sizes, int n_in,
                              void* d_out, int out_size, void* d_ws, size_t ws_size,
                              hipStream_t stream)
{
    (void)in_sizes; (void)n_in; (void)out_size; (void)ws_size;
    const float* x  = (const float*)d_in[0];
    const int*   ei = (const int*)d_in[1];
    const float* ea = (const float*)d_in[2];
    auto F = [&](int i) { return (const float*)d_in[i]; };

    // workspace carve-up (floats)
    const size_t SZ = (size_t)NNODES * HIDD;   // 12.8M floats
    float* ws    = (float*)d_ws;
    float* qb    = ws;
    float* kb    = ws + SZ;
    float* vb    = ws + 2 * SZ;
    float* ob    = ws + 3 * SZ;                // skip + aggregated attention
    float* hb    = ws + 4 * SZ;                // layer activations
    float* alpha = ws + 5 * SZ;                // NEDGES*4
    float* denom = alpha + (size_t)NEDGES * 4; // NNODES*4
    int*   amax  = (int*)(denom + (size_t)NNODES * 4);
    float* colsum = (float*)(amax + (size_t)NNODES * 4);
    float* colsq  = colsum + HIDD;
    float* t1     = colsq + HIDD;              // 2000x128
    float* t2     = t1 + (size_t)NJOBS * 128;  // 2000x64

    const int eh = NEDGES * 4;
    const int nh = NNODES * 4;

    auto run_conv = [&](const float* hin, int K, int pb, int bnb) {
        wmma_gemm4_lds<<<NNODES / 16, 256, 0, stream>>>(
            hin, F(pb + 0), F(pb + 1), F(pb + 2), F(pb + 3), F(pb + 4), F(pb + 5),
            F(pb + 7), F(pb + 8), qb, kb, vb, ob, K);
        fill_i32<<<(nh + 255) / 256, 256, 0, stream>>>(amax, 0x80000000, nh);
        fill_i32<<<(nh + 255) / 256, 256, 0, stream>>>((int*)denom, 0, nh);
        edge_alpha_kernel<<<(eh + 255) / 256, 256, 0, stream>>>(ei, ea, F(pb + 6), qb, kb, alpha, amax);
        edge_ex_kernel<<<(eh + 255) / 256, 256, 0, stream>>>(ei, alpha, amax, denom);
        edge_agg_kernel<<<NEDGES, 256, 0, stream>>>(ei, ea, F(pb + 6), vb, alpha, denom, ob);
        fill_i32<<<2, 256, 0, stream>>>((int*)colsum, 0, 2 * HIDD);
        bn_reduce_kernel<<<256, 256, 0, stream>>>(ob, colsum, colsq, NNODES);
        bn_apply_kernel<<<(int)((SZ + 255) / 256), 256, 0, stream>>>(
            ob, colsum, colsq, F(bnb), F(bnb + 1), hb, NNODES);
    };

    // params flat order: conv1@3, conv2@12, conv3@21, bn1@30, bn2@32, bn3@34,
    // job@36(6), demand@42(4), hot@46(4)
    run_conv(x,  128, 3,  30);
    run_conv(hb, 256, 12, 32);
    run_conv(hb, 256, 21, 34);

    float* out = (float*)d_out;
    auto gblk = [](int M, int N) { return ((M / 16) * (N / 16) + 7) / 8; };

    // job head: 256 ->128(leaky) ->64(leaky) ->2000 logits
    wmma_gemm<<<gblk(NJOBS, 128), 256, 0, stream>>>(hb, F(36), F(37), t1, NJOBS, 128, 256, 1);
    wmma_gemm<<<gblk(NJOBS, 64),  256, 0, stream>>>(t1, F(38), F(39), t2, NJOBS, 64, 128, 1);
    wmma_gemm<<<gblk(NJOBS, 2000), 256, 0, stream>>>(t2, F(40), F(41), out, NJOBS, 2000, 64, 0);

    // demand head: 256 ->64(leaky) ->1, sigmoid
    wmma_gemm<<<gblk(NJOBS, 64), 256, 0, stream>>>(hb, F(42), F(43), t1, NJOBS, 64, 256, 1);
    dot_sigmoid_kernel<<<(NJOBS + 255) / 256, 256, 0, stream>>>(t1, F(44), F(45),
                                                                out + (size_t)NJOBS * NJOBS, NJOBS, 16);
    // hot head
    wmma_gemm<<<gblk(NJOBS, 64), 256, 0, stream>>>(hb, F(46), F(47), t1, NJOBS, 64, 256, 1);
    dot_sigmoid_kernel<<<(NJOBS + 255) / 256, 256, 0, stream>>>(t1, F(48), F(49),
                                                                out + (size_t)NJOBS * NJOBS + NJOBS, NJOBS, 16);
}